// GetFilter_66305705115959
// MI455X (gfx1250) — compile-verified
//
#include <hip/hip_runtime.h>

// ---------------------------------------------------------------------------
// ChebNet layer for MI455X (gfx1250, wave32, WMMA).
//   N=16 batches, M=2048 nodes, FIN=16, K=6, FOUT=128
//   T0 = x ; T1 = L x ; Tk = 2 L T_{k-1} - T_{k-2}
//   out = relu(concat_k(Tk) @ W + B)
// Dominant cost: 5 streaming passes over L (256 MB) -> fp32 WMMA 16x16x4.
// Round 2: unroll x16 for deeper VMEM pipelining + register-pipelined
// double-buffered LDS staging of the T operand (1 barrier per chunk).
// ---------------------------------------------------------------------------

typedef float v2f __attribute__((ext_vector_type(2)));
typedef float v4f __attribute__((ext_vector_type(4)));
typedef float v8f __attribute__((ext_vector_type(8)));

#define N_B   16
#define M_N   2048
#define FIN   16
#define KORD  6
#define FOUT  128
#define FTOT  (FIN * KORD)   // 96
#define CHUNK 512            // LDS-staged rows of T per pass (32 KB per buffer)
#define NCHUNK (M_N / CHUNK) // 4

// ---------------------------------------------------------------------------
// Kernel 1: scatter x (=T0) into feat[:, f*6 + 0]
// ---------------------------------------------------------------------------
__global__ void cheb_pack_x_kernel(const float* __restrict__ x,
                                   float* __restrict__ feat) {
    int gid = blockIdx.x * blockDim.x + threadIdx.x;   // 0 .. N*M*FIN-1
    if (gid >= N_B * M_N * FIN) return;
    int f  = gid & 15;
    int nm = gid >> 4;
    feat[(size_t)nm * FTOT + f * KORD + 0] = x[gid];
}

// ---------------------------------------------------------------------------
// Kernel 2: one Chebyshev step.
//   Tout = alpha * (L @ Tprev) + beta * Tprev2      (per batch)
//   also writes Tout into feat[:, f*6 + kstep]
// Grid: (16 row-blocks of 128 rows, 16 batches), block = 256 threads (8 waves)
// Each wave computes a 16x16 tile: 16 rows of L times 16 feature columns,
// reduction depth 2048 via 512 x v_wmma_f32_16x16x4_f32.
// T operand double-buffered in LDS; next chunk prefetched into registers
// while the current chunk computes.
// ---------------------------------------------------------------------------
__global__ __launch_bounds__(256)
void cheb_step_kernel(const float* __restrict__ Lm,      // [N][M][M]
                      const float* __restrict__ Tprev,   // [N][M][FIN]
                      const float* __restrict__ Tprev2,  // [N][M][FIN]
                      float* __restrict__ Tout,          // [N][M][FIN]
                      float* __restrict__ feat,          // [N*M][FTOT]
                      int kstep, float alpha, float beta) {
    __shared__ float ldsT[2 * CHUNK * FIN];              // 2 x 32 KB

    const int tid  = threadIdx.x;
    const int n    = blockIdx.y;                         // batch
    const int rb   = blockIdx.x;                         // 128-row block
    const int wave = tid >> 5;
    const int lane = tid & 31;
    const int hi   = lane >> 4;                          // half-wave select
    const int lidx = lane & 15;
    const int koff = hi * 2;                             // K sub-offset (ISA A/B layout)

    const int m0   = rb * 128 + wave * 16;               // this wave's row tile
    const int arow = m0 + lidx;                          // A-fragment row (M = lane&15)
    const float* Lrow = Lm + ((size_t)n * M_N + arow) * (size_t)M_N;
    const float* Tsrc = Tprev + (size_t)n * M_N * FIN;

    // ---- stage chunk 0 into LDS buffer 0 ----
    v4f stage[8];                                        // 32 VGPRs of staging data
    {
        const v4f* src = (const v4f*)Tsrc;
        #pragma unroll
        for (int i = 0; i < 8; ++i) stage[i] = src[tid + i * 256];
        v4f* dst = (v4f*)ldsT;
        #pragma unroll
        for (int i = 0; i < 8; ++i) dst[tid + i * 256] = stage[i];
    }
    __syncthreads();

    v8f acc = {0.f, 0.f, 0.f, 0.f, 0.f, 0.f, 0.f, 0.f};

    for (int c = 0; c < NCHUNK; ++c) {
        // ---- prefetch next chunk into registers (overlaps with compute) ----
        if (c < NCHUNK - 1) {
            const v4f* src = (const v4f*)(Tsrc + (size_t)(c + 1) * CHUNK * FIN);
            #pragma unroll
            for (int i = 0; i < 8; ++i) stage[i] = src[tid + i * 256];
        }

        const float* Tl = ldsT + (c & 1) * (CHUNK * FIN);
        const float* Lc = Lrow + c * CHUNK;

        // ---- reduction over this chunk: 128 WMMA steps of K=4, unroll 16 ----
        #pragma unroll 16
        for (int kk = 0; kk < CHUNK; kk += 4) {
            // A 16x4 fp32 fragment: lane&15 = row, lanes 0-15 hold K=kk,kk+1,
            // lanes 16-31 hold K=kk+2,kk+3 -> contiguous float2 per lane.
            v2f a = *(const v2f*)(Lc + kk + koff);
            // B 4x16 fp32 fragment: lane&15 = column, same K split.
            v2f b;
            b.x = Tl[(kk + koff) * FIN + lidx];
            b.y = Tl[(kk + koff + 1) * FIN + lidx];
            acc = __builtin_amdgcn_wmma_f32_16x16x4_f32(
                false, a, false, b, (short)0, acc, false, false);
        }

        // ---- dump prefetched chunk into the other LDS buffer ----
        if (c < NCHUNK - 1) {
            v4f* dst = (v4f*)(ldsT + ((c + 1) & 1) * (CHUNK * FIN));
            #pragma unroll
            for (int i = 0; i < 8; ++i) dst[tid + i * 256] = stage[i];
        }
        __syncthreads();
    }

    // ---- epilogue: recurrence combine + dual store (T layout + feat layout) ----
    #pragma unroll
    for (int v = 0; v < 8; ++v) {
        int   mrow = m0 + v + hi * 8;                    // C layout: M = v (+8 for hi half)
        size_t idx = ((size_t)n * M_N + mrow) * FIN + lidx;
        float val  = alpha * acc[v] + beta * Tprev2[idx];
        Tout[idx] = val;
        feat[((size_t)n * M_N + mrow) * FTOT + lidx * KORD + kstep] = val;
    }
}

// ---------------------------------------------------------------------------
// Kernel 3: out = relu(feat @ W + bias)
//   feat [32768 x 96], W [96 x 128], out [32768 x 128]
// One wave per 16-row tile; 8 WMMA accumulators cover all 128 output cols.
// ---------------------------------------------------------------------------
__global__ __launch_bounds__(256)
void cheb_head_kernel(const float* __restrict__ feat,
                      const float* __restrict__ W,
                      const float* __restrict__ bias,
                      float* __restrict__ out) {
    const int tid  = threadIdx.x;
    const int wave = tid >> 5;
    const int lane = tid & 31;
    const int hi   = lane >> 4;
    const int lidx = lane & 15;
    const int koff = hi * 2;
    const int r0   = (blockIdx.x * 8 + wave) * 16;       // row tile of feat

    const float* arow = feat + (size_t)(r0 + lidx) * FTOT;

    v8f acc[8];
    #pragma unroll
    for (int t = 0; t < 8; ++t)
        acc[t] = (v8f){0.f, 0.f, 0.f, 0.f, 0.f, 0.f, 0.f, 0.f};

    #pragma unroll
    for (int kk = 0; kk < FTOT; kk += 4) {
        v2f a = *(const v2f*)(arow + kk + koff);         // contiguous b64
        const float* w0 = W + (size_t)(kk + koff) * FOUT + lidx;
        #pragma unroll
        for (int t = 0; t < 8; ++t) {
            v2f b;
            b.x = w0[t * 16];
            b.y = w0[FOUT + t * 16];
            acc[t] = __builtin_amdgcn_wmma_f32_16x16x4_f32(
                false, a, false, b, (short)0, acc[t], false, false);
        }
    }

    #pragma unroll
    for (int t = 0; t < 8; ++t) {
        float bv = bias[t * 16 + lidx];
        #pragma unroll
        for (int v = 0; v < 8; ++v) {
            int   r   = r0 + v + hi * 8;
            float val = acc[t][v] + bv;
            out[(size_t)r * FOUT + t * 16 + lidx] = fmaxf(val, 0.f);
        }
    }
}

// ---------------------------------------------------------------------------
// Host launcher
// ---------------------------------------------------------------------------
extern "C" void kernel_launch(void* const* d_in, const int* in_sizes, int n_in,
                              void* d_out, int out_size, void* d_ws, size_t ws_size,
                              hipStream_t stream) {
    const float* x  = (const float*)d_in[0];   // [16][2048][16]
    const float* Lm = (const float*)d_in[1];   // [16][2048][2048]
    const float* W  = (const float*)d_in[2];   // [96][128]
    const float* B  = (const float*)d_in[3];   // [128]
    float* out = (float*)d_out;                // [16][2048][128]

    // Workspace layout: feat [32768*96] then three T ping-pong buffers [N*M*FIN]
    float* feat = (float*)d_ws;
    float* T1 = feat + (size_t)N_B * M_N * FTOT;
    float* T2 = T1 + (size_t)N_B * M_N * FIN;
    float* T3 = T2 + (size_t)N_B * M_N * FIN;

    // T0 = x -> feat column k=0
    cheb_pack_x_kernel<<<(N_B * M_N * FIN + 255) / 256, 256, 0, stream>>>(x, feat);

    dim3 grid(16, N_B);
    // k=1: T1 = L x
    cheb_step_kernel<<<grid, 256, 0, stream>>>(Lm, x,  x,  T1, feat, 1, 1.0f,  0.0f);
    // k=2: T2 = 2 L T1 - x
    cheb_step_kernel<<<grid, 256, 0, stream>>>(Lm, T1, x,  T2, feat, 2, 2.0f, -1.0f);
    // k=3: T3 = 2 L T2 - T1
    cheb_step_kernel<<<grid, 256, 0, stream>>>(Lm, T2, T1, T3, feat, 3, 2.0f, -1.0f);
    // k=4: T4 = 2 L T3 - T2   (reuse T1 buffer)
    cheb_step_kernel<<<grid, 256, 0, stream>>>(Lm, T3, T2, T1, feat, 4, 2.0f, -1.0f);
    // k=5: T5 = 2 L T4 - T3   (reuse T2 buffer)
    cheb_step_kernel<<<grid, 256, 0, stream>>>(Lm, T1, T3, T2, feat, 5, 2.0f, -1.0f);

    // out = relu(feat @ W + B)
    cheb_head_kernel<<<256, 256, 0, stream>>>(feat, W, B, out);
}